// ARMAConv_56908316672633
// MI455X (gfx1250) — compile-verified
//
#include <hip/hip_runtime.h>

// ARMAConv (PyG) for MI455X / gfx1250 (wave32, WMMA).
// gcn_norm -> CSR build (no f32 atomics in hot path) ->
// [WMMA GEMM -> CSR-gather propagate -> fused root-GEMM+bias+relu] x2 -> mean over K.
// Dense 64x64 transforms: V_WMMA_F32_16X16X4_F32 (exact fp32).
// Propagate: gather-based, register accumulation, single store per output row chunk.
// Working set (~120 MB) fits MI455X 192 MB L2 -> replays run at L2 bandwidth.

typedef __attribute__((ext_vector_type(2))) float v2f;
typedef __attribute__((ext_vector_type(8))) float v8f;

#define KSTACK 2
#define FDIM   64

// ---------------- elementwise helpers ----------------

__global__ void k_zerof(float* __restrict__ p, long long n) {
  long long i = (long long)blockIdx.x * blockDim.x + threadIdx.x;
  if (i < n) p[i] = 0.0f;
}

__global__ void k_zeroi(int* __restrict__ p, long long n) {
  long long i = (long long)blockIdx.x * blockDim.x + threadIdx.x;
  if (i < n) p[i] = 0;
}

// deg[col[e]] += w[e]  (float, for gcn_norm) and cnt[col[e]] += 1 (int, for CSR).
__global__ void k_deg_hist(const int* __restrict__ col, const float* __restrict__ w,
                           float* __restrict__ deg, int* __restrict__ cnt, long long E) {
  long long e = (long long)blockIdx.x * blockDim.x + threadIdx.x;
  if (e < E) {
    int c = col[e];
    unsafeAtomicAdd(deg + c, w[e]);
    atomicAdd(cnt + c, 1);
  }
}

__global__ void k_gcnnorm(const int* __restrict__ row, const int* __restrict__ col,
                          const float* __restrict__ w, const float* __restrict__ deg,
                          float* __restrict__ norm, long long E) {
  long long e = (long long)blockIdx.x * blockDim.x + threadIdx.x;
  if (e >= E) return;
  float dr = deg[row[e]];
  float dc = deg[col[e]];
  float ir = dr > 0.f ? rsqrtf(fmaxf(dr, 1e-12f)) : 0.f;
  float ic = dc > 0.f ? rsqrtf(fmaxf(dc, 1e-12f)) : 0.f;
  norm[e] = ir * w[e] * ic;
}

// ---------------- CSR build: two-level exclusive scan + placement ----------------

// Per-256 block inclusive scan of cnt -> exclusive into startp, block total into bsum.
__global__ void k_scan_block(const int* __restrict__ cnt, int* __restrict__ startp,
                             int* __restrict__ bsum, int n) {
  __shared__ int s[256];
  const int tid = threadIdx.x;
  const int i = blockIdx.x * 256 + tid;
  const int v = (i < n) ? cnt[i] : 0;
  s[tid] = v;
  __syncthreads();
#pragma unroll
  for (int d = 1; d < 256; d <<= 1) {
    int t = (tid >= d) ? s[tid - d] : 0;
    __syncthreads();
    s[tid] += t;
    __syncthreads();
  }
  if (i < n) startp[i] = s[tid] - v;
  if (tid == 255) bsum[blockIdx.x] = s[255];
}

// Single-block exclusive scan of block sums (nb <= 512).
__global__ void k_scan_bsums(const int* __restrict__ bsum, int* __restrict__ boff, int nb) {
  __shared__ int s[512];
  const int tid = threadIdx.x;
  const int v = (tid < nb) ? bsum[tid] : 0;
  s[tid] = v;
  __syncthreads();
#pragma unroll
  for (int d = 1; d < 512; d <<= 1) {
    int t = (tid >= d) ? s[tid - d] : 0;
    __syncthreads();
    s[tid] += t;
    __syncthreads();
  }
  if (tid < nb) boff[tid] = s[tid] - v;
}

// start[i] += boff[i/256]; fill[i] = start[i] (fill becomes running cursor, then "end").
__global__ void k_add_off(int* __restrict__ startp, const int* __restrict__ boff,
                          int* __restrict__ fill, int n) {
  int i = blockIdx.x * blockDim.x + threadIdx.x;
  if (i < n) {
    int sv = startp[i] + boff[i >> 8];
    startp[i] = sv;
    fill[i] = sv;
  }
}

// Place edge e at pos within its destination bucket; copy src row + norm into CSR order.
__global__ void k_place(const int* __restrict__ row, const int* __restrict__ col,
                        const float* __restrict__ norm, int* __restrict__ fill,
                        int* __restrict__ srcp, float* __restrict__ nrmp, long long E) {
  long long e = (long long)blockIdx.x * blockDim.x + threadIdx.x;
  if (e < E) {
    int pos = atomicAdd(fill + col[e], 1);
    srcp[pos] = row[e];
    nrmp[pos] = norm[e];
  }
}

// ---------------- gather propagate: P[k,n,:] = sum_e norm * H[k,src,:] ----------------
// 16 threads per destination node, each owns one float4 column chunk; both k-planes
// accumulated in registers; exactly one store per (k, node, chunk). No atomics.
__global__ void k_gather(const float* __restrict__ H, const int* __restrict__ srcp,
                         const float* __restrict__ nrmp, const int* __restrict__ startp,
                         const int* __restrict__ endp, float* __restrict__ P, long long N) {
  long long gid = (long long)blockIdx.x * blockDim.x + threadIdx.x;
  if (gid >= N * 16) return;
  const int n = (int)(gid >> 4);
  const int c = (int)(gid & 15) * 4;
  const int b = startp[n];
  const int e = endp[n];
  const float* __restrict__ H0 = H + c;
  const float* __restrict__ H1 = H + (long long)N * FDIM + c;
  float4 a0 = make_float4(0.f, 0.f, 0.f, 0.f);
  float4 a1 = make_float4(0.f, 0.f, 0.f, 0.f);
  for (int i = b; i < e; ++i) {
    const int r = srcp[i];
    const float w = nrmp[i];
    const long long ro = (long long)r * FDIM;
    const float4 h0 = *(const float4*)(H0 + ro);
    const float4 h1 = *(const float4*)(H1 + ro);
    a0.x += w * h0.x; a0.y += w * h0.y; a0.z += w * h0.z; a0.w += w * h0.w;
    a1.x += w * h1.x; a1.y += w * h1.y; a1.z += w * h1.z; a1.w += w * h1.w;
  }
  const long long no = (long long)n * FDIM + c;
  *(float4*)(P + no) = a0;
  *(float4*)(P + (long long)N * FDIM + no) = a1;
}

// ---------------- WMMA GEMM: Y[k] = A[k] @ W[k] (+ optional epilogue) ----------------
// One wave = 16 rows x 64 cols; K consumed in 16 steps of 4 (v_wmma_f32_16x16x4_f32).
// f32 A-frag (16x4): lane half h, m = lane%16: vgpr v = A[m, 4kk + v + 2h].
// f32 B-frag (4x16): vgpr v = W[4kk + v + 2h, j*16 + n].
// f32 D (16x16): vgpr v = row (v + 8h), col = j*16 + n.
// All A staged in registers before any store -> in-place (Y==A) is safe per wave.
template <int MODE>
__global__ void k_gemm64(const float* __restrict__ A, long long a_kstride,
                         const float* __restrict__ W,
                         const float* __restrict__ P,
                         const float* __restrict__ bias,
                         float* __restrict__ Y, int n_rows) {
  const int wave = threadIdx.x >> 5;
  const int lane = threadIdx.x & 31;
  const int h    = lane >> 4;
  const int ln   = lane & 15;
  const int k    = blockIdx.y;

  const int row0 = (blockIdx.x * 8 + wave) * 16;
  if (row0 >= n_rows) return;  // uniform per wave

  const float* Ak = A + (long long)k * a_kstride;
  const float* Wk = W + (long long)k * (FDIM * FDIM);

  const int m  = row0 + ln;
  const int mc = m < n_rows ? m : n_rows - 1;  // clamped rows are never stored

  v2f areg[16];
  const float* arow = Ak + (long long)mc * FDIM + 2 * h;
#pragma unroll
  for (int kk = 0; kk < 16; ++kk) {
    areg[kk] = *(const v2f*)(arow + 4 * kk);
  }

  const bool full = (row0 + 16 <= n_rows);  // wave-uniform

#pragma unroll
  for (int j = 0; j < 4; ++j) {
    v8f acc = {};
#pragma unroll
    for (int kk = 0; kk < 16; ++kk) {
      const float* wp = Wk + (4 * kk + 2 * h) * FDIM + j * 16 + ln;
      v2f b;
      b.x = wp[0];
      b.y = wp[FDIM];
      acc = __builtin_amdgcn_wmma_f32_16x16x4_f32(
          false, areg[kk], false, b, (short)0, acc, false, false);
    }
    const int colc = j * 16 + ln;
    const long long base = ((long long)k * n_rows + row0 + 8 * h) * FDIM + colc;
    float* __restrict__ Yb = Y + base;
    if (full) {
      if (MODE == 1) {
        const float* __restrict__ Pb = P + base;
        const float bb = bias[k * FDIM + colc];
#pragma unroll
        for (int v = 0; v < 8; ++v) {
          float val = acc[v] + Pb[(long long)v * FDIM] + bb;
          Yb[(long long)v * FDIM] = val > 0.f ? val : 0.f;
        }
      } else {
#pragma unroll
        for (int v = 0; v < 8; ++v) Yb[(long long)v * FDIM] = acc[v];
      }
    } else {
      const float bb = (MODE == 1) ? bias[k * FDIM + colc] : 0.f;
#pragma unroll
      for (int v = 0; v < 8; ++v) {
        const int r = row0 + v + 8 * h;
        if (r < n_rows) {
          float val = acc[v];
          if (MODE == 1) {
            val += P[base + (long long)v * FDIM] + bb;
            val = val > 0.f ? val : 0.f;
          }
          Yb[(long long)v * FDIM] = val;
        }
      }
    }
  }
}

// out[n,f] = 0.5 * (H[0,n,f] + H[1,n,f])
__global__ void k_mean(const float4* __restrict__ H, float4* __restrict__ out,
                       long long n4, long long kstride4) {
  long long i = (long long)blockIdx.x * blockDim.x + threadIdx.x;
  if (i < n4) {
    float4 a = H[i];
    float4 b = H[i + kstride4];
    out[i] = make_float4(0.5f * (a.x + b.x), 0.5f * (a.y + b.y),
                         0.5f * (a.z + b.z), 0.5f * (a.w + b.w));
  }
}

// ---------------- host launch ----------------

extern "C" void kernel_launch(void* const* d_in, const int* in_sizes, int n_in,
                              void* d_out, int out_size, void* d_ws, size_t ws_size,
                              hipStream_t stream) {
  const float* x      = (const float*)d_in[0];  // [N, 64]
  const int*   eidx   = (const int*)d_in[1];    // [2, E]
  const float* ew     = (const float*)d_in[2];  // [E]
  const float* init_w = (const float*)d_in[3];  // [K, 64, 64]
  const float* weight = (const float*)d_in[4];  // [T-1, K, 64, 64]
  const float* root_w = (const float*)d_in[5];  // [T, K, 64, 64]
  const float* bias   = (const float*)d_in[6];  // [T, K, 64]
  float* out = (float*)d_out;                   // [N, 64]

  const long long N = (long long)in_sizes[0] / FDIM;
  const long long E = (long long)in_sizes[2];
  const int* row = eidx;       // edge_index[0] (src)
  const int* col = eidx + E;   // edge_index[1] (dst)

  const long long knf = (long long)KSTACK * N * FDIM;

  // Workspace layout (4-byte units), 16B-aligned blocks first:
  // H[K*N*64] | P[K*N*64] | deg[N] | norm[E] | nrmp[E] | cnt[N] | start[N] | fill[N]
  // | srcp[E] | bsum[512] | boff[512]
  float* ws    = (float*)d_ws;
  float* H     = ws;
  float* P     = H + knf;
  float* deg   = P + knf;
  float* norm  = deg + N;
  float* nrmp  = norm + E;
  int*   cnt   = (int*)(nrmp + E);
  int*   start = cnt + N;
  int*   fill  = start + N;
  int*   srcp  = fill + N;
  int*   bsum  = srcp + E;
  int*   boff  = bsum + 512;

  const int B = 256;
  auto nb = [](long long n) { return (unsigned)((n + 255) / 256); };
  const unsigned nblk = nb(N);  // scan blocks (<= 512 required; 391 for N=100k)
  const dim3 ggrid((unsigned)((N + 127) / 128), KSTACK);

  // --- gcn_norm + CSR build ---
  k_zerof<<<nb(N), B, 0, stream>>>(deg, N);
  k_zeroi<<<nb(N), B, 0, stream>>>(cnt, N);
  k_deg_hist<<<nb(E), B, 0, stream>>>(col, ew, deg, cnt, E);
  k_gcnnorm<<<nb(E), B, 0, stream>>>(row, col, ew, deg, norm, E);
  k_scan_block<<<nblk, 256, 0, stream>>>(cnt, start, bsum, (int)N);
  k_scan_bsums<<<1, 512, 0, stream>>>(bsum, boff, (int)nblk);
  k_add_off<<<nb(N), B, 0, stream>>>(start, boff, fill, (int)N);
  k_place<<<nb(E), B, 0, stream>>>(row, col, norm, fill, srcp, nrmp, E);
  // After placement: start[n] = begin, fill[n] = end of bucket n.

  // --- t = 0 ---
  k_gemm64<0><<<ggrid, B, 0, stream>>>(x, 0, init_w, nullptr, nullptr, H, (int)N);
  k_gather<<<nb(N * 16), B, 0, stream>>>(H, srcp, nrmp, start, fill, P, N);
  k_gemm64<1><<<ggrid, B, 0, stream>>>(x, 0, root_w, P, bias, H, (int)N);

  // --- t = 1 ---
  k_gemm64<0><<<ggrid, B, 0, stream>>>(H, N * FDIM, weight, nullptr, nullptr, H, (int)N);
  k_gather<<<nb(N * 16), B, 0, stream>>>(H, srcp, nrmp, start, fill, P, N);
  k_gemm64<1><<<ggrid, B, 0, stream>>>(x, 0, root_w + (long long)KSTACK * FDIM * FDIM,
                                       P, bias + KSTACK * FDIM, H, (int)N);

  // --- mean over K ---
  const long long nf4 = N * FDIM / 4;
  k_mean<<<nb(nf4), B, 0, stream>>>((const float4*)H, (float4*)out, nf4, nf4);
}